// Sgv1ToRGB_88210038325488
// MI455X (gfx1250) — compile-verified
//
#include <hip/hip_runtime.h>
#include <math.h>

// Problem constants (match reference)
#define B_   8
#define C_   128
#define S_   512
#define H_   256
#define W_   256
#define HW_  (H_ * W_)
#define OC_  3
#define SH_  (H_ / 2)
#define SW_  (W_ / 2)

#define SQRT2f      1.41421356237309515f
#define LIN_SCALE   0.04419417382415922f   // 1/sqrt(512)
#define CONV_SCALE  0.08838834764831845f   // 1/sqrt(128)

typedef float v2f __attribute__((ext_vector_type(2)));
typedef float v8f __attribute__((ext_vector_type(8)));

__device__ __forceinline__ float act_lrelu(float t) {
    // (x >= 0 ? x : 0.2x) * sqrt(2)
    return (t >= 0.0f ? t : 0.2f * t) * SQRT2f;
}

// ---------------------------------------------------------------------------
// K1: per-(b,c) plane statistics of activated input. One block per plane.
// Bandwidth-bound: streams 268 MB with float4 loads + speculative prefetch.
// ---------------------------------------------------------------------------
__global__ __launch_bounds__(256)
void stats_kernel(const float* __restrict__ input,
                  const float* __restrict__ act_bias,
                  float* __restrict__ meanv,
                  float* __restrict__ rsigv)
{
    const int plane = blockIdx.x;               // b*C + c
    const int c   = plane & (C_ - 1);
    const float ab = act_bias[c];
    const float4* src = (const float4*)(input + (size_t)plane * HW_);
    const int tid = threadIdx.x;

    float sum = 0.0f, sq = 0.0f;
    for (int i = tid; i < HW_ / 4; i += 256) {
        int pf = i + 512; if (pf >= HW_ / 4) pf = HW_ / 4 - 1;
        __builtin_prefetch((const void*)(src + pf), 0, 0);   // global_prefetch_b8
        float4 v = src[i];
        float a;
        a = act_lrelu(v.x + ab); sum += a; sq = fmaf(a, a, sq);
        a = act_lrelu(v.y + ab); sum += a; sq = fmaf(a, a, sq);
        a = act_lrelu(v.z + ab); sum += a; sq = fmaf(a, a, sq);
        a = act_lrelu(v.w + ab); sum += a; sq = fmaf(a, a, sq);
    }

    __shared__ float ssum[256];
    __shared__ float ssq[256];
    ssum[tid] = sum; ssq[tid] = sq;
    __syncthreads();
    for (int off = 128; off > 0; off >>= 1) {
        if (tid < off) { ssum[tid] += ssum[tid + off]; ssq[tid] += ssq[tid + off]; }
        __syncthreads();
    }
    if (tid == 0) {
        float m   = ssum[0] * (1.0f / HW_);
        float var = ssq[0]  * (1.0f / HW_) - m * m;
        var = fmaxf(var, 0.0f);
        meanv[plane] = m;
        rsigv[plane] = rsqrtf(var + 1e-4f);
    }
}

// ---------------------------------------------------------------------------
// K2: style modulation GEMM via V_WMMA_F32_16X16X4_F32 (exact f32 math).
//   gamma(16pad x 128) = style(16pad x 512) * mod_weight^T(512 x 128)
// One workgroup, 8 waves; wave w owns C-columns [16w, 16w+16). K=512 -> 128
// wmma issues per wave. Then folds rsig / conv-weight scaling into eff_w.
//
// Fragment layouts per ISA 7.12.2:
//   A (16x4, M x K):  lane<16: M=lane,    {VGPR0,VGPR1}={K0,K1}
//                     lane>=16: M=lane-16, {VGPR0,VGPR1}={K2,K3}
//   B (4x16, K x N):  lane index = N; half/VGPR select K (mirrors sparse-B
//                     half-lane pattern documented in 7.12.5)
//   C/D (16x16):      VGPR r, lane L -> M = r + 8*(L>=16), N = L&15
// ---------------------------------------------------------------------------
__global__ __launch_bounds__(256)
void modulate_wmma_kernel(const float* __restrict__ style,       // (B,S)
                          const float* __restrict__ mod_weight,  // (C,S)
                          const float* __restrict__ mod_bias,    // (C)
                          const float* __restrict__ conv_weight, // (OC,C)
                          const float* __restrict__ rsigv,       // (B,C)
                          float* __restrict__ eff_w)             // (B,OC,C)
{
    const int tid   = threadIdx.x;
    const int wave  = tid >> 5;          // 0..7 -> C tile
    const int lane  = tid & 31;
    const int hi    = lane >> 4;         // 0/1: which K half
    const int ln    = lane & 15;
    const int cbase = wave * 16;

    const int  m      = ln;              // A row (style batch), padded to 16
    const bool mvalid = (m < B_);
    const float* srow = style + (size_t)(mvalid ? m : 0) * S_;   // clamped: always in-bounds
    const float* wrow = mod_weight + (size_t)(cbase + ln) * S_;  // B column n -> mod_weight row

    v8f acc = {};
    for (int kk = 0; kk < S_; kk += 4) {
        const int ka = kk + (hi ? 2 : 0);
        v2f a, b;
        float a0 = srow[ka], a1 = srow[ka + 1];
        a.x = mvalid ? a0 : 0.0f;        // zero-pad batch rows 8..15
        a.y = mvalid ? a1 : 0.0f;
        b.x = wrow[ka];
        b.y = wrow[ka + 1];
        // EXEC is all-ones here (no divergent control flow around the wmma)
        acc = __builtin_amdgcn_wmma_f32_16x16x4_f32(
            /*neg_a=*/false, a, /*neg_b=*/false, b,
            /*c_mod=*/(short)0, acc, /*reuse_a=*/false, /*reuse_b=*/false);
    }

    const int   c     = cbase + ln;
    const float mbias = mod_bias[c];
#pragma unroll
    for (int r = 0; r < 8; ++r) {
        const int bi = r + hi * 8;       // output row M
        if (bi < B_) {
            float gam = fmaf(acc[r], LIN_SCALE, mbias);     // EqualLinear scale + bias
            float g   = gam * rsigv[bi * C_ + c];           // fold InstanceNorm sigma
#pragma unroll
            for (int o = 0; o < OC_; ++o)
                eff_w[(bi * OC_ + o) * C_ + c] =
                    conv_weight[o * C_ + c] * CONV_SCALE * g;
        }
    }
}

// ---------------------------------------------------------------------------
// K2b: eff_b[b,o] = conv_bias[o] + out_bias[o] - sum_c eff_w[b,o,c]*mean[b,c]
// ---------------------------------------------------------------------------
__global__ void effb_kernel(const float* __restrict__ eff_w,
                            const float* __restrict__ meanv,
                            const float* __restrict__ conv_bias,
                            const float* __restrict__ out_bias,
                            float* __restrict__ eff_b)
{
    const int t = threadIdx.x;
    if (t >= B_ * OC_) return;
    const int b = t / OC_, o = t % OC_;
    float s = 0.0f;
    for (int c = 0; c < C_; ++c)
        s = fmaf(eff_w[(b * OC_ + o) * C_ + c], meanv[b * C_ + c], s);
    eff_b[t] = conv_bias[o] + out_bias[o] - s;
}

__device__ __forceinline__ float skip_at(const float* __restrict__ sp, int sy, int sx) {
    return (sy >= 0 && sy < SH_ && sx >= 0 && sx < SW_) ? sp[sy * SW_ + sx] : 0.0f;
}

// ---------------------------------------------------------------------------
// K3: streaming pass. Each thread owns 4 consecutive pixels (float4);
// loops over all 128 channels (coalesced 4KB/block/channel), accumulates the
// 3 effective-conv outputs, then adds eff_b and the separable [1,3,3,1]/4
// FIR 2x upsample of skip, and stores float4.
// ---------------------------------------------------------------------------
__global__ __launch_bounds__(256)
void fused_main_kernel(const float* __restrict__ input,
                       const float* __restrict__ act_bias,
                       const float* __restrict__ skip,
                       const float* __restrict__ eff_w,
                       const float* __restrict__ eff_b,
                       float* __restrict__ out)
{
    const int b   = blockIdx.y;
    const int tid = threadIdx.x;
    const int p0  = (blockIdx.x * 256 + tid) * 4;

    __shared__ float s_w[OC_ * C_];
    __shared__ float s_b[OC_];
    __shared__ float s_ab[C_];
    for (int i = tid; i < OC_ * C_; i += 256) s_w[i] = eff_w[b * OC_ * C_ + i];
    for (int i = tid; i < C_; i += 256)       s_ab[i] = act_bias[i];
    if (tid < OC_)                            s_b[tid] = eff_b[b * OC_ + tid];
    __syncthreads();

    const float* base = input + (size_t)b * C_ * HW_ + p0;
    float4 a0 = {0,0,0,0}, a1 = {0,0,0,0}, a2 = {0,0,0,0};

    for (int c = 0; c < C_; ++c) {
        int pc = c + 4; if (pc >= C_) pc = C_ - 1;
        __builtin_prefetch((const void*)(base + (size_t)pc * HW_), 0, 0);
        float4 v = *(const float4*)(base + (size_t)c * HW_);
        const float ab = s_ab[c];
        float4 a;
        a.x = act_lrelu(v.x + ab);
        a.y = act_lrelu(v.y + ab);
        a.z = act_lrelu(v.z + ab);
        a.w = act_lrelu(v.w + ab);
        const float w0 = s_w[c], w1 = s_w[C_ + c], w2 = s_w[2 * C_ + c];
        a0.x = fmaf(w0, a.x, a0.x); a0.y = fmaf(w0, a.y, a0.y);
        a0.z = fmaf(w0, a.z, a0.z); a0.w = fmaf(w0, a.w, a0.w);
        a1.x = fmaf(w1, a.x, a1.x); a1.y = fmaf(w1, a.y, a1.y);
        a1.z = fmaf(w1, a.z, a1.z); a1.w = fmaf(w1, a.w, a1.w);
        a2.x = fmaf(w2, a.x, a2.x); a2.y = fmaf(w2, a.y, a2.y);
        a2.z = fmaf(w2, a.z, a2.z); a2.w = fmaf(w2, a.w, a2.w);
    }

    // --- separable FIR 2x upsample taps (upfirdn2d up=2, pad=(2,1), k=[1,3,3,1]) ---
    const int y = p0 >> 8;               // W_ == 256
    int syA, syB; float wyA, wyB;
    if ((y & 1) == 0) { syA = y >> 1;       wyA = 0.75f; syB = (y >> 1) - 1; wyB = 0.25f; }
    else              { syA = (y + 1) >> 1; wyA = 0.25f; syB = (y - 1) >> 1; wyB = 0.75f; }

    int sxA[4], sxB[4]; float wxA[4], wxB[4];
#pragma unroll
    for (int j = 0; j < 4; ++j) {
        const int x = (p0 & 255) + j;
        if ((x & 1) == 0) { sxA[j] = x >> 1;       wxA[j] = 0.75f; sxB[j] = (x >> 1) - 1; wxB[j] = 0.25f; }
        else              { sxA[j] = (x + 1) >> 1; wxA[j] = 0.25f; sxB[j] = (x - 1) >> 1; wxB[j] = 0.75f; }
    }

    float accs[OC_][4] = {
        { a0.x, a0.y, a0.z, a0.w },
        { a1.x, a1.y, a1.z, a1.w },
        { a2.x, a2.y, a2.z, a2.w } };

#pragma unroll
    for (int o = 0; o < OC_; ++o) {
        const float* sp = skip + (size_t)(b * OC_ + o) * (SH_ * SW_);
        const float eb = s_b[o];
        float r[4];
#pragma unroll
        for (int j = 0; j < 4; ++j) {
            float rowA = wxA[j] * skip_at(sp, syA, sxA[j]) + wxB[j] * skip_at(sp, syA, sxB[j]);
            float rowB = wxA[j] * skip_at(sp, syB, sxA[j]) + wxB[j] * skip_at(sp, syB, sxB[j]);
            r[j] = accs[o][j] + eb + (wyA * rowA + wyB * rowB);
        }
        float4 res = { r[0], r[1], r[2], r[3] };
        *(float4*)(out + (size_t)(b * OC_ + o) * HW_ + p0) = res;
    }
}

// ---------------------------------------------------------------------------
// Launch: stats -> wmma modulation -> eff_b -> fused streaming pass.
// Workspace layout (floats): mean[1024] | rsig[1024] | eff_w[3072] | eff_b[24]
// (~20.6 KB, all fully rewritten each call -> deterministic).
// ---------------------------------------------------------------------------
extern "C" void kernel_launch(void* const* d_in, const int* in_sizes, int n_in,
                              void* d_out, int out_size, void* d_ws, size_t ws_size,
                              hipStream_t stream)
{
    const float* input       = (const float*)d_in[0];
    const float* style       = (const float*)d_in[1];
    const float* skip        = (const float*)d_in[2];
    const float* act_bias    = (const float*)d_in[3];
    const float* mod_weight  = (const float*)d_in[4];
    const float* mod_bias    = (const float*)d_in[5];
    const float* conv_weight = (const float*)d_in[6];
    const float* conv_bias   = (const float*)d_in[7];
    const float* out_bias    = (const float*)d_in[8];
    float* out = (float*)d_out;

    float* ws    = (float*)d_ws;
    float* meanv = ws;                        // B*C
    float* rsigv = ws + B_ * C_;              // B*C
    float* eff_w = ws + 2 * B_ * C_;          // B*OC*C
    float* eff_b = eff_w + B_ * OC_ * C_;     // B*OC

    stats_kernel<<<B_ * C_, 256, 0, stream>>>(input, act_bias, meanv, rsigv);
    modulate_wmma_kernel<<<1, 256, 0, stream>>>(style, mod_weight, mod_bias,
                                                conv_weight, rsigv, eff_w);
    effb_kernel<<<1, 32, 0, stream>>>(eff_w, meanv, conv_bias, out_bias, eff_b);
    dim3 grid(HW_ / (256 * 4), B_);
    fused_main_kernel<<<grid, 256, 0, stream>>>(input, act_bias, skip,
                                                eff_w, eff_b, out);
}